// FilterDetection_63050119905767
// MI455X (gfx1250) — compile-verified
//
#include <hip/hip_runtime.h>
#include <cstdint>

// FilterDetection post-processing for MI455X (gfx1250, wave32).
// One workgroup per batch; sort + greedy NMS with early exit at 100 survivors.
// Uses the CDNA5 Tensor Data Mover to stage the proposal block (correctness
// does not depend on TDM semantics; destination is rewritten after the wait).

#define B_ 8
#define N_ 2048
#define C_ 81
#define P_ 100
#define T_ 1024
#define SENT_ 0xFF800000u   // key-hi for score == -inf (non-keep)

#if defined(__gfx1250__)
typedef unsigned int tdm_v4u __attribute__((ext_vector_type(4)));
typedef int          tdm_v8i __attribute__((ext_vector_type(8)));
typedef int          tdm_v4i __attribute__((ext_vector_type(4)));
#endif

__global__ __launch_bounds__(T_) void fd_kernel(const float* __restrict__ logit,
                                                const float* __restrict__ regress,
                                                const float* __restrict__ prop,
                                                float* __restrict__ out_logit,
                                                float* __restrict__ out_box) {
  __shared__ float              s_box[N_ * 4];        // 32 KB: decoded, clipped boxes
  __shared__ unsigned long long s_key[N_];            // 16 KB: sort keys (score desc, idx asc)
  __shared__ unsigned char     s_lab[N_];            //  2 KB: argmax class per proposal
  __shared__ unsigned char     s_keep[N_];           //  2 KB: per-rank survivor flag
  __shared__ int               s_surv[P_];           // surviving orig indices, rank order
  __shared__ int               s_nsurv, s_r, s_accept, s_stop, s_albl;
  __shared__ float             s_abox[4];

  const int tid = threadIdx.x;
  const int b   = blockIdx.x;

#if defined(__gfx1250__)
  // --- Tensor Data Mover: DMA this batch's proposal tile (8192 f32 = 32 KB)
  // into LDS (s_box reused as landing buffer; rewritten below). One wave issues.
  if (tid < 32) {
    unsigned long long ga = (unsigned long long)(uintptr_t)(prop + (size_t)b * N_ * 4);
    unsigned int       la = (unsigned int)(uintptr_t)(&s_box[0]);
    // D# group0: count=1 | lds_addr | global_addr[56:0] | type=2
    tdm_v4u g0 = { 1u, la, (unsigned int)ga,
                   (unsigned int)(((ga >> 32) & 0x01FFFFFFull) | 0x80000000ull) };
    // D# group1: data_size=4B; tensor_dim0=8192; tensor_dim1=1; tile=8192x1;
    //            tensor_dim0_stride=8192; no multicast/pad/iterate/barrier.
    tdm_v8i g1 = { (int)(2u << 16),
                   (int)(8192u << 16),
                   (int)(1u << 16),
                   (int)(8192u << 16),
                   1,
                   8192,
                   0, 0 };
    tdm_v4i z4 = { 0, 0, 0, 0 };
#if __has_include(<hip/amd_detail/amd_gfx1250_TDM.h>)
    tdm_v8i z8 = { 0, 0, 0, 0, 0, 0, 0, 0 };
    __builtin_amdgcn_tensor_load_to_lds(g0, g1, z4, z4, z8, 0);
#else
    __builtin_amdgcn_tensor_load_to_lds(g0, g1, z4, z4, 0);
#endif
    __builtin_amdgcn_s_wait_tensorcnt(0);
  }
#endif
  __syncthreads();

  // --- Phase 1: per-proposal argmax, box decode, sort key ------------------
  const float MAXR = 4.1351666f; // |log(16/1000)|
  for (int i = tid; i < N_; i += T_) {
    const size_t row = (size_t)b * N_ + i;
    const float4 pr  = reinterpret_cast<const float4*>(prop)[row];
    const float  mx  = fmaxf(fmaxf(pr.x, pr.y), fmaxf(pr.z, pr.w));

    const float* lrow = logit + row * C_;
    float best = lrow[0];
    int   lab  = 0;
    for (int c = 1; c < C_; ++c) {
      const float v = lrow[c];
      if (v > best) { best = v; lab = c; }   // first-occurrence argmax
    }

    const float* dp = regress + (row * C_ + lab) * 4;
    const float d0 = dp[0] * 0.1f, d1 = dp[1] * 0.1f;
    const float dw = fminf(fmaxf(dp[2] * 0.2f, -MAXR), MAXR);
    const float dh = fminf(fmaxf(dp[3] * 0.2f, -MAXR), MAXR);
    const float w  = pr.z - pr.x, h = pr.w - pr.y;
    const float cx = pr.x + 0.5f * w, cy = pr.y + 0.5f * h;
    const float ncx = cx + d0 * w, ncy = cy + d1 * h;
    const float nw  = w * expf(dw), nh = h * expf(dh);
    s_box[i * 4 + 0] = fminf(fmaxf(ncx - 0.5f * nw, 0.f), 1.f);
    s_box[i * 4 + 1] = fminf(fmaxf(ncy - 0.5f * nh, 0.f), 1.f);
    s_box[i * 4 + 2] = fminf(fmaxf(ncx + 0.5f * nw, 0.f), 1.f);
    s_box[i * 4 + 3] = fminf(fmaxf(ncy + 0.5f * nh, 0.f), 1.f);
    s_lab[i] = (unsigned char)lab;

    const bool keep = (mx > 0.f) && (lab > 0) && (best >= 0.7f);
    // order-preserving uint of score (score >= 0 when kept); hi = ~ord(score)
    const unsigned int u = keep ? (__float_as_uint(best) ^ 0x80000000u)
                                : 0x007FFFFFu;            // ord(-inf)
    s_key[i] = ((unsigned long long)(~u) << 32) | (unsigned int)i;
  }
  __syncthreads();

  // --- Phase 2: bitonic sort (ascending key == descending score, stable) ---
  for (int k = 2; k <= N_; k <<= 1) {
    for (int j = k >> 1; j > 0; j >>= 1) {
      for (int i = tid; i < N_; i += T_) {
        const int p = i ^ j;
        if (p > i) {
          const unsigned long long a = s_key[i], c = s_key[p];
          const bool up = ((i & k) == 0);
          if (up ? (a > c) : (a < c)) { s_key[i] = c; s_key[p] = a; }
        }
      }
      __syncthreads();
    }
  }

  for (int r = tid; r < N_; r += T_)
    s_keep[r] = ((unsigned int)(s_key[r] >> 32) != SENT_) ? 1 : 0;
  if (tid == 0) { s_nsurv = 0; s_r = 0; }
  __syncthreads();

  // --- Phase 3: greedy per-class NMS with early exit at P_ survivors -------
  while (true) {
    if (tid == 0) {
      int rr = s_r;
      int acc = 0;
      while (rr < N_) {
        if ((unsigned int)(s_key[rr] >> 32) == SENT_) { rr = N_; break; } // rest are -inf
        if (s_keep[rr]) break;
        ++rr;
      }
      if (rr < N_) {
        const int orig = (int)(unsigned int)s_key[rr];
        s_surv[s_nsurv++] = orig;
        acc = 1;
        s_abox[0] = s_box[orig * 4 + 0];
        s_abox[1] = s_box[orig * 4 + 1];
        s_abox[2] = s_box[orig * 4 + 2];
        s_abox[3] = s_box[orig * 4 + 3];
        s_albl = s_lab[orig];
        s_r = rr + 1;
        s_stop = (s_nsurv >= P_) ? 1 : 0;   // 100th found: later ranks irrelevant
      } else {
        s_r = rr;
        s_stop = 1;
      }
      s_accept = acc;
    }
    __syncthreads();
    if (s_stop) break;                      // uniform
    if (s_accept) {
      const float ax1 = s_abox[0], ay1 = s_abox[1], ax2 = s_abox[2], ay2 = s_abox[3];
      const int   al  = s_albl;
      const float areaA = fmaxf(ax2 - ax1, 0.f) * fmaxf(ay2 - ay1, 0.f);
      for (int j = s_r + tid; j < N_; j += T_) {
        if (!s_keep[j]) continue;
        const int oj = (int)(unsigned int)s_key[j];
        if ((int)s_lab[oj] != al) continue;
        const float bx1 = s_box[oj * 4 + 0], by1 = s_box[oj * 4 + 1];
        const float bx2 = s_box[oj * 4 + 2], by2 = s_box[oj * 4 + 3];
        const float areaB = fmaxf(bx2 - bx1, 0.f) * fmaxf(by2 - by1, 0.f);
        const float iw = fmaxf(fminf(ax2, bx2) - fmaxf(ax1, bx1), 0.f);
        const float ih = fmaxf(fminf(ay2, by2) - fmaxf(ay1, by1), 0.f);
        const float inter = iw * ih;
        const float iou = inter / fmaxf(areaA + areaB - inter, 1e-12f);
        if (iou > 0.3f) s_keep[j] = 0;
      }
    }
    __syncthreads();
  }
  __syncthreads();

  // --- Phase 4: write outputs (zero-padded past survivor count) ------------
  const int ns = s_nsurv;
  float* ol = out_logit + (size_t)b * P_ * C_;
  for (int e = tid; e < P_ * C_; e += T_) {
    const int p = e / C_, c = e - p * C_;
    float v = 0.f;
    if (p < ns) v = logit[((size_t)b * N_ + s_surv[p]) * C_ + c];
    ol[e] = v;
  }
  float* ob = out_box + (size_t)b * P_ * 4;
  for (int e = tid; e < P_ * 4; e += T_) {
    const int p = e >> 2, c = e & 3;
    float v = 0.f;
    if (p < ns) v = s_box[s_surv[p] * 4 + c];
    ob[e] = v;
  }
}

extern "C" void kernel_launch(void* const* d_in, const int* in_sizes, int n_in,
                              void* d_out, int out_size, void* d_ws, size_t ws_size,
                              hipStream_t stream) {
  (void)in_sizes; (void)n_in; (void)out_size; (void)d_ws; (void)ws_size;
  const float* logit   = (const float*)d_in[0];   // [8, 2048, 81]
  const float* regress = (const float*)d_in[1];   // [8, 2048, 81, 4]
  const float* prop    = (const float*)d_in[2];   // [8, 2048, 4]
  float* out_logit = (float*)d_out;                         // [8, 100, 81]
  float* out_box   = out_logit + (size_t)B_ * P_ * C_;      // [8, 100, 4]
  fd_kernel<<<dim3(B_), dim3(T_), 0, stream>>>(logit, regress, prop, out_logit, out_box);
}